// BiGraphContrastLayer_25409026523344
// MI455X (gfx1250) — compile-verified
//
#include <hip/hip_runtime.h>
#include <hip/hip_bf16.h>
#include <climits>

// ---------------------------------------------------------------------------
// GATConv forward for MI455X (gfx1250, wave32).
//   N=100000 nodes, E=1.6M edges (+N self loops), IN=256, HEADS=8, HD=16.
// Phase 1: h = feat @ W via V_WMMA_F32_16X16X4_F32 (exact fp32, matrix pipe).
// Phase 2: per-node logits el/er, 3-pass edge softmax with atomics.
// Phase 3: bandwidth-bound weighted scatter-add (dominant cost ~75us HBM).
// ---------------------------------------------------------------------------

#define IN_DIM   256
#define OUT_DIM  128
#define HEADS    8
#define HEAD_DIM 16
#define NEG_SLOPE 0.2f

typedef float v2f __attribute__((ext_vector_type(2)));
typedef float v8f __attribute__((ext_vector_type(8)));

// Monotonic float <-> int key for integer atomicMax (total order incl. negatives).
__device__ __forceinline__ int floatKey(float x) {
    int i = __float_as_int(x);
    return (i >= 0) ? i : (i ^ 0x7FFFFFFF);
}
__device__ __forceinline__ float keyToFloat(int k) {
    int i = (k >= 0) ? k : (k ^ 0x7FFFFFFF);
    return __int_as_float(i);
}

__device__ __forceinline__ float lrelu(float x) {
    return x > 0.0f ? x : NEG_SLOPE * x;
}

// ---------------------------------------------------------------------------
// K0: init out = bias (broadcast), emax keys = -inf, denom = 0.
// ---------------------------------------------------------------------------
__global__ void gat_init(float* __restrict__ out, const float* __restrict__ b,
                         int* __restrict__ emaxk, float* __restrict__ denom, int N) {
    int gid = blockIdx.x * blockDim.x + threadIdx.x;
    int tot = N * OUT_DIM;
    if (gid < tot) {
        out[gid] = b[gid & (OUT_DIM - 1)];
    }
    if (gid < N * HEADS) {
        emaxk[gid] = INT_MIN;
        denom[gid] = 0.0f;
    }
}

// ---------------------------------------------------------------------------
// K1: h = feat @ W  (fp32 WMMA 16x16x4).
// Block = 256 threads = 8 waves; block handles 16 rows, wave w handles head w
// (columns [16w,16w+16)). A tile (16x256) staged in LDS once, shared by all
// 8 waves; stride padded to 260 floats -> lanes m=0..15 hit distinct banks
// (codegen: ds_load_2addr_b64 per K-step pair).
//
// WMMA fp32 operand layout (ISA 7.12.2):
//   A 16x4 : lane L holds A[M=L%16][K = 2*(L/16)+j] in v[j], j=0,1
//   B 4x16 : lane L holds B[K = 2*(L/16)+j][N = L%16] in v[j]
//   C/D    : lane L, v[r] = C[M = r + 8*(L/16)][N = L%16], r=0..7
// ---------------------------------------------------------------------------
#define A_STRIDE 260  // 256 + 4 pad (keeps 16B alignment, banks m*4+k distinct)

__global__ void gat_gemm_wmma(const float* __restrict__ feat,
                              const float* __restrict__ W,
                              float* __restrict__ h, int N) {
    __shared__ float As[16 * A_STRIDE];

    const int row0 = blockIdx.x * 16;
    const int lane = threadIdx.x & 31;
    const int wave = threadIdx.x >> 5;          // head / column tile
    const int m    = lane & 15;                 // row-in-tile (A), col-in-tile (B)
    const int kh   = (lane >> 4) << 1;          // 0 or 2: K half selector

    // Cooperative A-tile stage: 16 rows x 256 cols as float4 (1024 vectors).
    for (int i = threadIdx.x; i < 16 * (IN_DIM / 4); i += blockDim.x) {
        int r = i >> 6;                          // /64 float4s per row
        int c = (i & 63) << 2;
        int gr = row0 + r;
        if (gr >= N) gr = N - 1;                 // clamp (keeps EXEC all-1 later)
        *(float4*)&As[r * A_STRIDE + c] =
            *(const float4*)&feat[(size_t)gr * IN_DIM + c];
    }
    __syncthreads();

    const int col0 = wave * HEAD_DIM;
    v8f acc = {};
    const float* __restrict__ arow = &As[m * A_STRIDE];

    #pragma unroll 4
    for (int k0 = 0; k0 < IN_DIM; k0 += 4) {
        v2f a = *(const v2f*)&arow[k0 + kh];               // A[m][k0+kh .. +1]
        v2f bb;
        bb.x = W[(size_t)(k0 + kh + 0) * OUT_DIM + col0 + m];
        bb.y = W[(size_t)(k0 + kh + 1) * OUT_DIM + col0 + m];
        acc = __builtin_amdgcn_wmma_f32_16x16x4_f32(
            /*neg_a=*/false, a, /*neg_b=*/false, bb,
            /*c_mod=*/(short)0, acc, /*reuse_a=*/false, /*reuse_b=*/false);
    }

    // Store D: lane holds column (col0+m), rows rbase..rbase+7.
    // Fast path for full tiles (all but possibly the last block): unguarded
    // stores with a common base -> compiler emits a clause of 8
    // global_store_b32 with immediate offsets (r*512B), no exec juggling.
    const int colg  = col0 + m;
    const int rbase = row0 + ((lane >> 4) << 3);
    float* __restrict__ op = h + (size_t)rbase * OUT_DIM + colg;
    if (row0 + 16 <= N) {
        #pragma unroll
        for (int r = 0; r < 8; ++r) {
            op[(size_t)r * OUT_DIM] = acc[r];
        }
    } else {
        #pragma unroll
        for (int r = 0; r < 8; ++r) {
            if (rbase + r < N) op[(size_t)r * OUT_DIM] = acc[r];
        }
    }
}

// ---------------------------------------------------------------------------
// K2: per-(node,head) attention logits el/er.
// ---------------------------------------------------------------------------
__global__ void gat_logits(const float* __restrict__ h,
                           const float* __restrict__ attn_l,
                           const float* __restrict__ attn_r,
                           float* __restrict__ el, float* __restrict__ er, int N) {
    int gid = blockIdx.x * blockDim.x + threadIdx.x;
    if (gid >= N * HEADS) return;
    int n  = gid >> 3;
    int hd = gid & 7;
    const float* hp = h + (size_t)n * OUT_DIM + hd * HEAD_DIM;
    const float* al = attn_l + hd * HEAD_DIM;
    const float* ar = attn_r + hd * HEAD_DIM;
    float sl = 0.0f, sr = 0.0f;
    #pragma unroll
    for (int j = 0; j < HEAD_DIM; ++j) {
        float v = hp[j];
        sl = fmaf(v, al[j], sl);
        sr = fmaf(v, ar[j], sr);
    }
    el[gid] = sl;
    er[gid] = sr;
}

// Fetch (s,d) for logical edge e; e >= E are the appended self loops.
__device__ __forceinline__ void edge_sd(const int* __restrict__ src,
                                        const int* __restrict__ dst,
                                        int e, int E, int& s, int& d) {
    if (e < E) { s = src[e]; d = dst[e]; }
    else       { s = e - E;  d = e - E; }
}

// ---------------------------------------------------------------------------
// K3: segment max of leaky_relu(el[s]+er[d]) over dst, via int-key atomicMax.
// ---------------------------------------------------------------------------
__global__ void gat_edge_max(const int* __restrict__ src, const int* __restrict__ dst,
                             const float* __restrict__ el, const float* __restrict__ er,
                             int* __restrict__ emaxk, int E, int Et) {
    int e = blockIdx.x * blockDim.x + threadIdx.x;
    if (e >= Et) return;
    int s, d;
    edge_sd(src, dst, e, E, s, d);
    float elv[8], erv[8];
    *(float4*)&elv[0] = *(const float4*)&el[(size_t)s * HEADS];
    *(float4*)&elv[4] = *(const float4*)&el[(size_t)s * HEADS + 4];
    *(float4*)&erv[0] = *(const float4*)&er[(size_t)d * HEADS];
    *(float4*)&erv[4] = *(const float4*)&er[(size_t)d * HEADS + 4];
    #pragma unroll
    for (int hd = 0; hd < HEADS; ++hd) {
        float x = lrelu(elv[hd] + erv[hd]);
        atomicMax(&emaxk[(size_t)d * HEADS + hd], floatKey(x));
    }
}

// ---------------------------------------------------------------------------
// K4: denom[d][h] = segment_sum exp(e - emax[d][h]).
// ---------------------------------------------------------------------------
__global__ void gat_edge_sum(const int* __restrict__ src, const int* __restrict__ dst,
                             const float* __restrict__ el, const float* __restrict__ er,
                             const int* __restrict__ emaxk, float* __restrict__ denom,
                             int E, int Et) {
    int e = blockIdx.x * blockDim.x + threadIdx.x;
    if (e >= Et) return;
    int s, d;
    edge_sd(src, dst, e, E, s, d);
    float elv[8], erv[8];
    *(float4*)&elv[0] = *(const float4*)&el[(size_t)s * HEADS];
    *(float4*)&elv[4] = *(const float4*)&el[(size_t)s * HEADS + 4];
    *(float4*)&erv[0] = *(const float4*)&er[(size_t)d * HEADS];
    *(float4*)&erv[4] = *(const float4*)&er[(size_t)d * HEADS + 4];
    #pragma unroll
    for (int hd = 0; hd < HEADS; ++hd) {
        float x  = lrelu(elv[hd] + erv[hd]);
        float mx = keyToFloat(emaxk[(size_t)d * HEADS + hd]);
        atomicAdd(&denom[(size_t)d * HEADS + hd], __expf(x - mx));
    }
}

// ---------------------------------------------------------------------------
// K5: out[d] += h[s] * alpha.  One thread per (edge, head): 64B contiguous
// gather of h[s], 16 fp32 atomic adds. Dominant HBM-bound phase.
// ---------------------------------------------------------------------------
__global__ void gat_aggregate(const int* __restrict__ src, const int* __restrict__ dst,
                              const float* __restrict__ el, const float* __restrict__ er,
                              const int* __restrict__ emaxk, const float* __restrict__ denom,
                              const float* __restrict__ h, float* __restrict__ out,
                              int E, int Et) {
    int gid = blockIdx.x * blockDim.x + threadIdx.x;
    if (gid >= Et * HEADS) return;
    int e  = gid >> 3;
    int hd = gid & 7;
    int s, d;
    edge_sd(src, dst, e, E, s, d);

    float x  = lrelu(el[(size_t)s * HEADS + hd] + er[(size_t)d * HEADS + hd]);
    float mx = keyToFloat(emaxk[(size_t)d * HEADS + hd]);
    float alpha = __expf(x - mx) / denom[(size_t)d * HEADS + hd];

    const float* hp = h + (size_t)s * OUT_DIM + hd * HEAD_DIM;
    float* op       = out + (size_t)d * OUT_DIM + hd * HEAD_DIM;
    #pragma unroll
    for (int v = 0; v < HEAD_DIM / 4; ++v) {
        float4 hv = *(const float4*)&hp[v * 4];
        atomicAdd(&op[v * 4 + 0], hv.x * alpha);
        atomicAdd(&op[v * 4 + 1], hv.y * alpha);
        atomicAdd(&op[v * 4 + 2], hv.z * alpha);
        atomicAdd(&op[v * 4 + 3], hv.w * alpha);
    }
}

// ---------------------------------------------------------------------------
// Host-side launcher.
// Inputs (setup_inputs order): feat(N*256), W(256*128), b(128),
//   attn_l(8*16), attn_r(8*16), src(E), dst(E).  Output: (N,128) fp32.
// ---------------------------------------------------------------------------
extern "C" void kernel_launch(void* const* d_in, const int* in_sizes, int n_in,
                              void* d_out, int out_size, void* d_ws, size_t ws_size,
                              hipStream_t stream) {
    const float* feat   = (const float*)d_in[0];
    const float* W      = (const float*)d_in[1];
    const float* b      = (const float*)d_in[2];
    const float* attn_l = (const float*)d_in[3];
    const float* attn_r = (const float*)d_in[4];
    const int*   src    = (const int*)d_in[5];
    const int*   dst    = (const int*)d_in[6];
    float*       out    = (float*)d_out;

    const int N  = in_sizes[0] / IN_DIM;
    const int E  = in_sizes[5];
    const int Et = E + N;                     // with self loops

    // Workspace carve-up: h | el | er | emax keys | denom  (~64 MB total)
    float* h     = (float*)d_ws;
    float* el    = h + (size_t)N * OUT_DIM;
    float* er    = el + (size_t)N * HEADS;
    int*   emaxk = (int*)(er + (size_t)N * HEADS);
    float* denom = (float*)(emaxk + (size_t)N * HEADS);

    const int TB = 256;

    gat_init<<<(N * OUT_DIM + TB - 1) / TB, TB, 0, stream>>>(out, b, emaxk, denom, N);

    gat_gemm_wmma<<<(N + 15) / 16, TB, 0, stream>>>(feat, W, h, N);

    gat_logits<<<(N * HEADS + TB - 1) / TB, TB, 0, stream>>>(h, attn_l, attn_r, el, er, N);

    gat_edge_max<<<(Et + TB - 1) / TB, TB, 0, stream>>>(src, dst, el, er, emaxk, E, Et);

    gat_edge_sum<<<(Et + TB - 1) / TB, TB, 0, stream>>>(src, dst, el, er, emaxk, denom, E, Et);

    gat_aggregate<<<((size_t)Et * HEADS + TB - 1) / TB, TB, 0, stream>>>(
        src, dst, el, er, emaxk, denom, h, out, E, Et);
}